// EmbeddingModel_10780367913066
// MI455X (gfx1250) — compile-verified
//
#include <hip/hip_runtime.h>
#include <stdint.h>

// SGNS (word2vec) loss for MI455X / gfx1250, wave32.
// One wave per batch element; dots via V_WMMA_F32_16X16X4_F32.
// Context-row gathers: GLOBAL_LOAD_ASYNC_TO_LDS_B128 (SADDR form, ASYNCcnt),
// double-buffered in LDS so tile t+1's 16 gathers overlap tile t's WMMAs.
// All gather address generation on the scalar path (v_readlane + SALU).

typedef __attribute__((ext_vector_type(2))) float v2f;
typedef __attribute__((ext_vector_type(8))) float v8f;

#define WPB  3     // waves per block (LDS: 3 * (2*8KB + 512B) = 50.7 KB)
#define DD   128   // embed dim
#define PP   10    // positives
#define NN   50    // negatives
#define CTX  60    // PP + NN
#define TILE 16
#define NTILES 4   // 64 rows (4 pad rows masked)

__global__ __launch_bounds__(WPB * 32)
void sgns_loss_wmma(const float* __restrict__ in_embed,
                    const float* __restrict__ out_embed,
                    const int*   __restrict__ input_labels,
                    const int*   __restrict__ pos_labels,
                    const int*   __restrict__ neg_labels,
                    float*       __restrict__ out,
                    int B)
{
    // per-wave private LDS: two 16x128 ctx tile buffers + 128-float input row
    __shared__ __align__(16) float lds[WPB][2 * TILE * DD + DD];

    const int lane = threadIdx.x & 31;
    const int wave = threadIdx.x >> 5;
    const int b    = blockIdx.x * WPB + wave;
    if (b >= B) return;   // whole-wave uniform exit; EXEC all-1 inside

    float* ctx0 = &lds[wave][0];
    float* ctx1 = &lds[wave][TILE * DD];
    float* vin  = &lds[wave][2 * TILE * DD];

    // ---- stage input embedding row (128 f32, float4 per lane) ----
    {
        const int ii = input_labels[b];
        const float4 v = ((const float4*)(in_embed + (size_t)ii * DD))[lane];
        ((float4*)vin)[lane] = v;
    }

    // ---- per-lane context labels for slots [lane] and [lane+32] ----
    const int s0 = lane;
    const int labLo = (s0 < PP) ? pos_labels[b * PP + s0]
                                : neg_labels[b * NN + (s0 - PP)];
    const int s1 = lane + 32;
    const int labHi = (s1 < CTX) ? neg_labels[b * NN + (s1 - PP)]
                                 : neg_labels[b * NN + (NN - 1)]; // pad rows (masked)

    const int half      = lane >> 4;     // 0: rows M=r, 1: rows M=r+8
    const int rowInHalf = lane & 15;
    const int koff      = 2 * half;      // A/B lane-half K offset per ISA layout

    const uint32_t voff = (uint32_t)(lane * 16);   // per-lane 16B chunk of a 512B row
    const uint64_t obase = (uint64_t)(uintptr_t)out_embed;

    // 16 async row-gathers of one tile; scalar base per row, SADDR addressing.
    auto issue_tile = [&](int t, float* dstbuf) {
        const uint32_t ldsbase = (uint32_t)(uintptr_t)dstbuf + voff; // LDS off = flat[31:0]
        #pragma unroll
        for (int r = 0; r < TILE; ++r) {
            const int slot = t * TILE + r;                 // compile-time constant
            const int idx  = (slot < 32)
                               ? __builtin_amdgcn_readlane(labLo, slot)
                               : __builtin_amdgcn_readlane(labHi, slot - 32);
            const uint64_t gbase = obase + ((uint64_t)(uint32_t)idx << 9); // idx*512B (SALU)
            const uint32_t la    = ldsbase + (uint32_t)(r * DD * 4);
            asm volatile("global_load_async_to_lds_b128 %0, %1, %2"
                         :: "v"(la), "v"(voff), "s"(gbase) : "memory");
        }
    };

    float part = 0.0f;

    issue_tile(0, ctx0);               // tile 0 gathers in flight

    #pragma unroll
    for (int t = 0; t < NTILES; ++t) {
        float* cur = (t & 1) ? ctx1 : ctx0;

        if (t + 1 < NTILES) {
            issue_tile(t + 1, (t & 1) ? ctx0 : ctx1); // next tile flies during WMMAs
            asm volatile("s_wait_asynccnt 0x10" ::: "memory"); // <=16 left => tile t done
        } else {
            asm volatile("s_wait_asynccnt 0x0" ::: "memory");
        }

        // ---- 16x16x4 f32 WMMA accumulation over K = 128 (2 chains) ----
        v8f c0 = {}, c1 = {};
        #pragma unroll
        for (int k = 0; k < DD; k += 8) {
            const int kk = k + koff;
            {
                v2f a  = *(const v2f*)(cur + rowInHalf * DD + kk);
                v2f bb = *(const v2f*)(vin + kk);
                c0 = __builtin_amdgcn_wmma_f32_16x16x4_f32(
                        false, a, false, bb, (short)0, c0, false, false);
            }
            {
                v2f a  = *(const v2f*)(cur + rowInHalf * DD + kk + 4);
                v2f bb = *(const v2f*)(vin + kk + 4);
                c1 = __builtin_amdgcn_wmma_f32_16x16x4_f32(
                        false, a, false, bb, (short)0, c1, false, false);
            }
        }

        // ---- log-sigmoid + masked partial sum ----
        // lane half 0 holds rows t*16 + r, half 1 holds rows t*16 + 8 + r
        #pragma unroll
        for (int r = 0; r < 8; ++r) {
            const int m   = t * TILE + 8 * half + r;
            const float d = c0[r] + c1[r];
            const float x = (m < PP) ? d : -d;                    // neg uses -inp
            const float ls = fminf(x, 0.0f) - __logf(1.0f + __expf(-fabsf(x)));
            part += (m < CTX) ? ls : 0.0f;
        }
    }

    // columns are identical within each 16-lane half; combine the two halves
    const float total = __shfl(part, 0, 32) + __shfl(part, 16, 32);
    if (lane == 0) out[b] = -total;
}

extern "C" void kernel_launch(void* const* d_in, const int* in_sizes, int n_in,
                              void* d_out, int out_size, void* d_ws, size_t ws_size,
                              hipStream_t stream)
{
    const float* in_embed     = (const float*)d_in[0];
    const float* out_embed    = (const float*)d_in[1];
    const int*   input_labels = (const int*)d_in[2];
    const int*   pos_labels   = (const int*)d_in[3];
    const int*   neg_labels   = (const int*)d_in[4];
    float* out = (float*)d_out;

    const int B = in_sizes[2];                       // 16384
    const int blocks = (B + WPB - 1) / WPB;

    sgns_loss_wmma<<<blocks, WPB * 32, 0, stream>>>(
        in_embed, out_embed, input_labels, pos_labels, neg_labels, out, B);
}